// ScRRAMBLeMLPMixerNetwork_19164144074965
// MI455X (gfx1250) — compile-verified
//
#include <hip/hip_runtime.h>
#include <math.h>
#include <stdint.h>

typedef __attribute__((ext_vector_type(2))) float v2f;
typedef __attribute__((ext_vector_type(8))) float v8f;
typedef unsigned int u32x4 __attribute__((ext_vector_type(4)));
typedef int i32x4 __attribute__((ext_vector_type(4)));
typedef int i32x8 __attribute__((ext_vector_type(8)));

#define BATCH 512
#define CORES 128
#define SI    4
#define SLOT  64
#define CS    256      /* SI*SLOT features per core */
#define FDIM  32768    /* CORES*CS */
#define NCOL  10

static __device__ __forceinline__ v8f wmma4(v2f a, v2f b, v8f c) {
  // D(16x16,f32) = A(16x4,f32) x B(4x16,f32) + C
  return __builtin_amdgcn_wmma_f32_16x16x4_f32(
      false, a, false, b, (short)0, c, false, false);
}

static __device__ __forceinline__ float warp_sum(float v) {
#pragma unroll
  for (int off = 16; off > 0; off >>= 1) v += __shfl_xor(v, off, 32);
  return v;
}

static __device__ __forceinline__ float gelu_tanh(float v) {
  float v3 = v * v * v;
  return 0.5f * v * (1.0f + tanhf(0.7978845608028654f * (v + 0.044715f * v3)));
}

// ---------------------------------------------------------------------------
// Patching (rank-1 factorization of the literal einsum):
//   T[k*4+l] = sum_{i,j} C_patch[i,j,k,l]
// ---------------------------------------------------------------------------
__global__ void k_patch_T(const float* __restrict__ Cp, float* __restrict__ T) {
  const int t = threadIdx.x;  // 0..511
  float acc = 0.f;
#pragma unroll 4
  for (int r = 0; r < 192; ++r) acc += Cp[r * 512 + t];
  T[t] = acc;
}

//   S[b,e] = sum_{p1,p2,c} x[b, p1*8+h, p2*8+w, c],  e = h*8+w
//   H[b, core, slot, e] = S[b,e] * T[core*4+slot]
__global__ void k_patch_expand(const float* __restrict__ x,
                               const float* __restrict__ T,
                               float* __restrict__ H) {
  __shared__ float sS[64];
  __shared__ float sT[512];
  const int tid = threadIdx.x;  // 256 threads
  const int b = blockIdx.x;
  if (tid < 64) {
    const int h = tid >> 3, w = tid & 7;
    const float* xb = x + (size_t)b * (64 * 64 * 3);
    float acc = 0.f;
    for (int p1 = 0; p1 < 8; ++p1)
      for (int p2 = 0; p2 < 8; ++p2) {
        const float* px = xb + ((p1 * 8 + h) * 64 + (p2 * 8 + w)) * 3;
        acc += px[0] + px[1] + px[2];
      }
    sS[tid] = acc;
  }
  for (int t = tid; t < 512; t += 256) sT[t] = T[t];
  __syncthreads();
  float* Hb = H + (size_t)b * FDIM;
  for (int idx = tid; idx < FDIM; idx += 256)
    Hb[idx] = sS[idx & 63] * sT[idx >> 6];
}

// ---------------------------------------------------------------------------
// GEMM1 (per-core token/slot mixer):
//   Z[b, i, l2*64+j] = sum_{m,k} Wi[l][i,j,k,l2,m] * H[b, i, m*64+k]
// Per core i: out(512x256) = in(512x256) @ B(256x256),
//   B[in,o] = Wi[i, o&63, in&63, o>>6, in>>6]
// Grid: (128 cores, 8 batch blocks of 64). Block 256 = 8 waves.
// Wave tile: 32 rows x 64 cols -> 2 A-frags x 4 B-frags, 8 accumulators.
// ~192 B of cache traffic per 2048-FLOP WMMA.
// ---------------------------------------------------------------------------
__global__ void k_gemm1(const float* __restrict__ H, const float* __restrict__ Wl,
                        float* __restrict__ Z) {
  const int core = blockIdx.x;
  const int tid = threadIdx.x;
  const int wave = tid >> 5, lane = tid & 31;
  const int hi = (lane >= 16) ? 1 : 0;
  const int l16 = lane & 15;
  const int row0 = blockIdx.y * 64 + (wave & 1) * 32;  // 32-row wave slab
  const int cgrp = wave >> 1;                          // 4 col tiles: cgrp*64..+63

  const float* Arow0 = H + (size_t)(row0 + l16) * FDIM + core * CS;
  const float* Arow1 = Arow0 + (size_t)16 * FDIM;
  const float* Wc = Wl + (size_t)core * 65536;

  int wb[4];
#pragma unroll
  for (int t = 0; t < 4; ++t) {
    const int o = (cgrp * 4 + t) * 16 + l16;
    wb[t] = (o & 63) * 1024 + (o >> 6) * 4;
  }

  const v8f zero = {};
  v8f acc[2][4];
#pragma unroll
  for (int r = 0; r < 2; ++r)
#pragma unroll
    for (int t = 0; t < 4; ++t) acc[r][t] = zero;

  for (int kk = 0; kk < CS; kk += 4) {
    const int d = kk + (hi ? 2 : 0);
    const v2f a0 = *reinterpret_cast<const v2f*>(Arow0 + d);
    const v2f a1 = *reinterpret_cast<const v2f*>(Arow1 + d);
    const int boff = (d & 63) * 16 + (d >> 6);
    __builtin_prefetch(&Wc[wb[0] + boff + 1024]);  // global_prefetch_b8
#pragma unroll
    for (int t = 0; t < 4; ++t) {
      v2f bf;
      bf.x = Wc[wb[t] + boff];
      bf.y = Wc[wb[t] + boff + 16];
      acc[0][t] = wmma4(a0, bf, acc[0][t]);
      acc[1][t] = wmma4(a1, bf, acc[1][t]);
    }
  }

#pragma unroll
  for (int r2 = 0; r2 < 2; ++r2)
#pragma unroll
    for (int t = 0; t < 4; ++t) {
      const int o0 = (cgrp * 4 + t) * 16;
#pragma unroll
      for (int r = 0; r < 8; ++r) {
        const int row = row0 + r2 * 16 + r + (hi ? 8 : 0);
        Z[(size_t)row * FDIM + core * CS + o0 + l16] = acc[r2][t][r];
      }
    }
}

// ---------------------------------------------------------------------------
// LayerNorm over F=32768 per batch row (optionally fused gelu on input).
// ---------------------------------------------------------------------------
__global__ void k_layernorm(const float* __restrict__ in, float* __restrict__ out,
                            const float* __restrict__ scale,
                            const float* __restrict__ bias, int do_gelu) {
  __shared__ float redS[32], redQ[32];
  __shared__ float s_stats[2];
  const int tid = threadIdx.x;
  const size_t base = (size_t)blockIdx.x * FDIM;
  float vals[32];
  float lsum = 0.f, lsq = 0.f;
#pragma unroll
  for (int i = 0; i < 32; ++i) {
    float v = in[base + tid + i * 1024];
    if (do_gelu) v = gelu_tanh(v);
    vals[i] = v;
    lsum += v;
    lsq += v * v;
  }
  lsum = warp_sum(lsum);
  lsq = warp_sum(lsq);
  const int wave = tid >> 5, lane = tid & 31;
  if (lane == 0) { redS[wave] = lsum; redQ[wave] = lsq; }
  __syncthreads();
  if (tid < 32) {
    float a = warp_sum(redS[tid]);
    float q = warp_sum(redQ[tid]);
    if (tid == 0) {
      const float mu = a * (1.0f / FDIM);
      const float var = q * (1.0f / FDIM) - mu * mu;
      s_stats[0] = mu;
      s_stats[1] = rsqrtf(var + 1e-6f);
    }
  }
  __syncthreads();
  const float mu = s_stats[0], rs = s_stats[1];
#pragma unroll
  for (int i = 0; i < 32; ++i) {
    const int f = tid + i * 1024;
    out[base + f] = (vals[i] - mu) * rs * scale[f] + bias[f];
  }
}

// ---------------------------------------------------------------------------
// GEMM2 (core scrambling) + residual:
//   Z2[b,P,m] = H[b,P,m] + sum_Q CoT[P,Q] * Zg[b,Q,m]
//   CoT[P,Q] = Co[Q>>2, Q&3, P>>2, P&3]  (P,Q in 0..511, m in 0..63)
// Grid (512 b, 2 m-halves). The 512x32 activation panel (64 KB) is staged
// into LDS by ONE Tensor-Data-Mover descriptor (tile 32x512, row stride 64),
// waited on with s_wait_tensorcnt. Each wave fixes one 16-col m-tile, holds
// 8 accumulators over 8 P-tiles, and reuses every LDS B-fragment 8x.
// ---------------------------------------------------------------------------
__global__ void k_gemm2(const float* __restrict__ Zg, const float* __restrict__ Col,
                        const float* __restrict__ Hin, float* __restrict__ Z2) {
  __shared__ float sZ[512 * 32];  // 64 KB panel: Zg[b][0..511][half*32 .. +32)
  const int b = blockIdx.x;
  const int half = blockIdx.y;  // which 32 of the 64 m-columns
  const int tid = threadIdx.x;
  const int wave = tid >> 5, lane = tid & 31;
  const int hi = (lane >= 16) ? 1 : 0;
  const int l16 = lane & 15;
  const size_t zbase = (size_t)b * FDIM;

  if (tid < 32) {  // wave 0 issues the TDM load (EXEC is ignored by TDM)
    const uint64_t gaddr =
        (uint64_t)(uintptr_t)(Zg + zbase + half * 32);  // tile start (bytes)
    const unsigned lds_addr = (unsigned)(uintptr_t)(void*)sZ;
    u32x4 g0;
    g0.x = 1u;                                   // count=1 (valid descriptor)
    g0.y = lds_addr;                             // D#.lds_addr
    g0.z = (unsigned)gaddr;                      // global_addr[31:0]
    g0.w = (unsigned)(gaddr >> 32) | 0x80000000u;  // addr[56:32] | type=2<<30
    i32x8 g1;
    g1[0] = 0x20000;        // data_size=2 (4 bytes); no mask/pad/iterate
    g1[1] = 0;              // tensor_dim0[15:0]<<16 (dim0 = 1<<30 -> low16=0)
    g1[2] = 0x4000;         // tensor_dim0[31:16] | tensor_dim1[15:0]<<16
    g1[3] = 0x4000 | (32 << 16);   // tensor_dim1[31:16] | tile_dim0=32
    g1[4] = 512;            // tile_dim1=512 | tile_dim2=0
    g1[5] = 64;             // tensor_dim0_stride = 64 elements
    g1[6] = 0;
    g1[7] = 0;
    i32x4 gz = {};
#if defined(__clang_major__) && __clang_major__ >= 23
    i32x8 gz8 = {};
    __builtin_amdgcn_tensor_load_to_lds(g0, g1, gz, gz, gz8, 0);
#else
    __builtin_amdgcn_tensor_load_to_lds(g0, g1, gz, gz, 0);
#endif
    __builtin_amdgcn_s_wait_tensorcnt(0);
  }
  __syncthreads();

  const int mt = wave & 1;          // m-tile within this half
  const int mcol = mt * 16 + l16;   // LDS column
  const int m0g = half * 32 + mt * 16;
  const int pgroup = wave >> 1;     // 0..3 -> P-tiles pgroup + 4*s

  int ab[8];
#pragma unroll
  for (int s = 0; s < 8; ++s) {
    const int P = (pgroup + 4 * s) * 16 + l16;
    ab[s] = ((P >> 2) * 4 + (P & 3)) + (hi ? 1024 : 0);
  }

  const v8f zero = {};
  v8f acc[8];
#pragma unroll
  for (int s = 0; s < 8; ++s) acc[s] = zero;

  for (int k4 = 0; k4 < 512; k4 += 4) {
    const int q = k4 + (hi ? 2 : 0);
    v2f bf;                               // B-frag from LDS, reused 8x
    bf.x = sZ[q * 32 + mcol];
    bf.y = sZ[q * 32 + 32 + mcol];
    const int koff = (k4 >> 2) * 2048;
#pragma unroll
    for (int s = 0; s < 8; ++s) {
      v2f a;
      a.x = Col[ab[s] + koff];
      a.y = Col[ab[s] + koff + 512];
      acc[s] = wmma4(a, bf, acc[s]);
    }
  }

#pragma unroll
  for (int s = 0; s < 8; ++s) {
    const int pbase = (pgroup + 4 * s) * 16;
#pragma unroll
    for (int r = 0; r < 8; ++r) {
      const int Pr = pbase + r + (hi ? 8 : 0);
      const size_t idx = zbase + (size_t)Pr * 64 + m0g;
      Z2[idx + l16] = acc[s][r] + Hin[idx + l16];
    }
  }
}

// ---------------------------------------------------------------------------
// Classifier: logits(512x10) = H(512x32768) @ W_cls(32768x10) + b_cls
// ---------------------------------------------------------------------------
__global__ void k_classifier(const float* __restrict__ Hf, const float* __restrict__ Wc,
                             const float* __restrict__ bc, float* __restrict__ out) {
  __shared__ float part[8 * 256];
  const int b0 = blockIdx.x * 16;
  const int tid = threadIdx.x;
  const int wave = tid >> 5, lane = tid & 31;
  const int hi = (lane >= 16) ? 1 : 0;
  const int l16 = lane & 15;
  const float* Arow = Hf + (size_t)(b0 + l16) * FDIM;
  const bool valid = l16 < NCOL;
  v8f acc = {};
  const int kbeg = wave * 4096;
  for (int kk = kbeg; kk < kbeg + 4096; kk += 4) {
    const v2f a = *reinterpret_cast<const v2f*>(Arow + kk + (hi ? 2 : 0));
    const int q0 = kk + (hi ? 2 : 0);
    v2f bf;
    bf.x = valid ? Wc[(size_t)q0 * NCOL + l16] : 0.f;
    bf.y = valid ? Wc[(size_t)(q0 + 1) * NCOL + l16] : 0.f;
    acc = wmma4(a, bf, acc);
  }
#pragma unroll
  for (int r = 0; r < 8; ++r) {
    const int M = r + (hi ? 8 : 0);
    part[wave * 256 + M * 16 + l16] = acc[r];
  }
  __syncthreads();
  {
    float s = 0.f;
#pragma unroll
    for (int w = 0; w < 8; ++w) s += part[w * 256 + tid];
    const int M = tid >> 4, N = tid & 15;
    if (N < NCOL) out[(size_t)(b0 + M) * NCOL + N] = s + bc[N];
  }
}

// ---------------------------------------------------------------------------
extern "C" void kernel_launch(void* const* d_in, const int* in_sizes, int n_in,
                              void* d_out, int out_size, void* d_ws, size_t ws_size,
                              hipStream_t stream) {
  const float* x    = (const float*)d_in[0];
  const float* Cp   = (const float*)d_in[1];
  const float* Wi   = (const float*)d_in[2];
  const float* Co   = (const float*)d_in[3];
  const float* blnS = (const float*)d_in[4];
  const float* blnB = (const float*)d_in[5];
  const float* olnS = (const float*)d_in[6];
  const float* olnB = (const float*)d_in[7];
  const float* Wcls = (const float*)d_in[8];
  const float* bcls = (const float*)d_in[9];
  float* out = (float*)d_out;

  char* ws = (char*)d_ws;
  const size_t ACT = (size_t)BATCH * FDIM * sizeof(float);  // 64 MB
  float* T  = (float*)(ws);
  float* H  = (float*)(ws + 4096);
  float* Z  = (float*)(ws + 4096 + ACT);
  float* Z2 = (float*)(ws + 4096 + 2 * ACT);

  k_patch_T<<<1, 512, 0, stream>>>(Cp, T);
  k_patch_expand<<<BATCH, 256, 0, stream>>>(x, T, H);

  for (int l = 0; l < 3; ++l) {
    const float* Wl = Wi + (size_t)l * CORES * 65536;
    const float* Cl = Co + (size_t)l * CORES * SI * CORES * SI;
    k_gemm1<<<dim3(CORES, BATCH / 64), 256, 0, stream>>>(H, Wl, Z);
    k_layernorm<<<BATCH, 1024, 0, stream>>>(Z, Z, blnS + (size_t)l * FDIM,
                                            blnB + (size_t)l * FDIM, 1);
    k_gemm2<<<dim3(BATCH, 2), 256, 0, stream>>>(Z, Cl, H, Z2);
    k_layernorm<<<BATCH, 1024, 0, stream>>>(Z2, H, olnS + (size_t)l * FDIM,
                                            olnB + (size_t)l * FDIM, 0);
  }
  k_classifier<<<BATCH / 16, 256, 0, stream>>>(H, Wcls, bcls, out);
}